// LocationSensitiveAttention_48395691491911
// MI455X (gfx1250) — compile-verified
//
#include <hip/hip_runtime.h>
#include <hip/hip_bf16.h>

typedef float v2f __attribute__((ext_vector_type(2)));
typedef float v8f __attribute__((ext_vector_type(8)));

static constexpr int Bk     = 64;
static constexpr int Tk     = 1000;
static constexpr int DVk    = 512;
static constexpr int DQk    = 1024;
static constexpr int UNITSk = 128;
static constexpr int Fk     = 32;
static constexpr int KSk    = 31;
static constexpr int ROWS   = 32;          // t rows per block (2 WMMA M-tiles)
static constexpr int VROW   = DVk + 4;     // padded LDS row stride (dwords)
static constexpr int SSTR   = 1024;        // score workspace stride per batch

// ---------------------------------------------------------------------------
// Kernel 1: q[b,u] = query[b,:] @ Wq[:,u] + bq[u] + bm[u]   (bm folded in)
// ---------------------------------------------------------------------------
__global__ void lsa_qproj(const float* __restrict__ query,
                          const float* __restrict__ Wq,
                          const float* __restrict__ bq,
                          const float* __restrict__ bm,
                          float* __restrict__ qws) {
    const int b = blockIdx.x;
    const int u = threadIdx.x;             // 128 threads
    const float* qrow = query + (size_t)b * DQk;
    float acc = bq[u] + bm[u];
#pragma unroll 4
    for (int k = 0; k < DQk; ++k)
        acc = fmaf(qrow[k], Wq[k * UNITSk + u], acc);
    qws[b * UNITSk + u] = acc;
}

// ---------------------------------------------------------------------------
// Kernel 2 (WMMA): fused  m = values@Wm ;  loc = tanh(conv1d(prev_w)@Wl) ;
//                  score[b,t] = tanh(m + q + bm + loc) @ Wv + bv
// One block = (batch b, 32 t-rows). 256 threads = 8 waves; wave w owns units
// [16w, 16w+16). Each wave keeps TWO 16x16 f32 accumulators (t-rows 0..15 and
// 16..31 of the block) so every Wm B-fragment is reused by two
// v_wmma_f32_16x16x4_f32 ops -> halved L2 traffic on Wm, 2x WMMA ILP.
// ---------------------------------------------------------------------------
__global__ void __launch_bounds__(256)
lsa_score(const float* __restrict__ values,
          const float* __restrict__ prevw,
          const float* __restrict__ Wm,
          const float* __restrict__ Wv,
          const float* __restrict__ bv,
          const float* __restrict__ Wconv,   // [31][1][32] -> k*32+f
          const float* __restrict__ Wl,      // [32][128]   -> f*128+u
          const float* __restrict__ qws,     // [64][128] (q + bq + bm)
          float* __restrict__ score) {       // [64][SSTR]
    __shared__ float vals[ROWS * VROW];      // 32 x 516 dwords = 66 KB
    __shared__ float convo[ROWS * Fk];       // 32 x 32
    __shared__ float pw[64];                 // prev_w window (62 used)
    __shared__ float wpart[8 * ROWS];        // per-wave score partials

    const int tid  = threadIdx.x;
    const int t0   = blockIdx.x * ROWS;
    const int bb   = blockIdx.y;

    // ---- stage values[bb, t0:t0+32, :] into LDS (float4, coalesced) ----
    {
        const float4* gsrc = (const float4*)(values + ((size_t)bb * Tk) * DVk);
#pragma unroll
        for (int i = 0; i < 16; ++i) {
            int idx = i * 256 + tid;            // 4096 float4 total
            int r   = idx >> 7;                 // / (DV/4)
            int c4  = idx & 127;
            int t   = t0 + r;
            float4 v = make_float4(0.f, 0.f, 0.f, 0.f);
            if (t < Tk) v = gsrc[(size_t)t * (DVk / 4) + c4];
            *(float4*)&vals[r * VROW + c4 * 4] = v;
        }
    }
    // ---- stage prev_w window [t0-15, t0+46] ----
    if (tid < ROWS + KSk - 1) {
        int t = t0 - (KSk / 2) + tid;
        pw[tid] = (t >= 0 && t < Tk) ? prevw[(size_t)bb * Tk + t] : 0.f;
    }
    __syncthreads();

    // ---- conv1d: 32 rows x 32 filters (SAME padding, no bias) ----
#pragma unroll
    for (int j = tid; j < ROWS * Fk; j += 256) {
        int r = j >> 5, f = j & 31;
        float acc = 0.f;
#pragma unroll
        for (int k = 0; k < KSk; ++k)
            acc = fmaf(pw[r + k], Wconv[k * Fk + f], acc);
        convo[j] = acc;
    }
    __syncthreads();

    // ---- WMMA GEMM: two 16(t) x 16(u) tiles per wave, K = 512 ----
    const int wave = tid >> 5;
    const int lane = tid & 31;
    const int half = lane >> 4;
    const int row  = lane & 15;
    const int u    = wave * 16 + row;          // this lane's unit column (N)

    v8f acc0 = {};
    v8f acc1 = {};
#pragma unroll 8
    for (int k0 = 0; k0 < DVk; k0 += 4) {
        const int k = k0 + 2 * half;
        v2f a0 = *(const v2f*)&vals[row * VROW + k];          // tile 0 rows
        v2f a1 = *(const v2f*)&vals[(row + 16) * VROW + k];   // tile 1 rows
        v2f bm2;
        bm2.x = Wm[(size_t)k * UNITSk + u];                   // B[K=k  ][N=u]
        bm2.y = Wm[(size_t)(k + 1) * UNITSk + u];             // B[K=k+1][N=u]
        acc0 = __builtin_amdgcn_wmma_f32_16x16x4_f32(
            false, a0, false, bm2, (short)0, acc0, false, false);
        acc1 = __builtin_amdgcn_wmma_f32_16x16x4_f32(
            false, a1, false, bm2, (short)0, acc1, false, false);
    }

    // ---- epilogue: +q(+bq+bm), +tanh(conv@Wl), tanh, *Wv, reduce ----
    const float base = qws[bb * UNITSk + u];
    const float wv   = Wv[u];
    float part[16];
#pragma unroll
    for (int tt = 0; tt < 2; ++tt) {
#pragma unroll
        for (int v = 0; v < 8; ++v) {
            const int m = tt * 16 + 8 * half + v;   // t-row within block
            float l = 0.f;
#pragma unroll
            for (int f = 0; f < Fk; ++f)
                l = fmaf(convo[m * Fk + f], Wl[f * UNITSk + u], l);
            const float mv = (tt == 0) ? acc0[v] : acc1[v];
            const float x  = mv + base + tanhf(l);
            part[tt * 8 + v] = tanhf(x) * wv;
        }
    }
    // butterfly reduce across the 16 lanes of this half (xor stays in-group)
#pragma unroll
    for (int i = 0; i < 16; ++i) {
#pragma unroll
        for (int off = 1; off < 16; off <<= 1)
            part[i] += __shfl_xor(part[i], off, 32);
    }
    if (row == 0) {
#pragma unroll
        for (int tt = 0; tt < 2; ++tt)
#pragma unroll
            for (int v = 0; v < 8; ++v)
                wpart[wave * ROWS + tt * 16 + 8 * half + v] = part[tt * 8 + v];
    }
    __syncthreads();

    if (tid < ROWS) {
        const int t = t0 + tid;
        float s = bv[0];
#pragma unroll
        for (int w = 0; w < 8; ++w) s += wpart[w * ROWS + tid];
        if (t < Tk) score[bb * SSTR + t] = s;
    }
}

// ---------------------------------------------------------------------------
// Kernel 3: softmax over T, write attention weights, context = aw^T @ values
// One block per batch; 512 threads (one DV column each in the context phase).
// ---------------------------------------------------------------------------
__global__ void __launch_bounds__(512)
lsa_softmax_ctx(const float* __restrict__ score,
                const float* __restrict__ values,
                float* __restrict__ out_ctx,    // [64][512]
                float* __restrict__ out_aw) {   // [64][1000]
    __shared__ float red[512];
    __shared__ float p[Tk];

    const int b   = blockIdx.x;
    const int tid = threadIdx.x;
    const float* srow = score + b * SSTR;

    float lmax = -3.0e38f;
    for (int t = tid; t < Tk; t += 512) lmax = fmaxf(lmax, srow[t]);
    red[tid] = lmax;
    __syncthreads();
#pragma unroll
    for (int s = 256; s > 0; s >>= 1) {
        if (tid < s) red[tid] = fmaxf(red[tid], red[tid + s]);
        __syncthreads();
    }
    const float mx = red[0];
    __syncthreads();

    float lsum = 0.f;
    for (int t = tid; t < Tk; t += 512) {
        float e = expf(srow[t] - mx);
        p[t] = e;
        lsum += e;
    }
    red[tid] = lsum;
    __syncthreads();
#pragma unroll
    for (int s = 256; s > 0; s >>= 1) {
        if (tid < s) red[tid] += red[tid + s];
        __syncthreads();
    }
    const float inv = 1.0f / red[0];
    __syncthreads();

    for (int t = tid; t < Tk; t += 512) {
        float a = p[t] * inv;
        p[t] = a;
        out_aw[(size_t)b * Tk + t] = a;
    }
    __syncthreads();

    // context: thread d accumulates over T (coalesced across threads)
    const int d = tid;
    const float* vb = values + (size_t)b * Tk * DVk + d;
    float acc = 0.f;
#pragma unroll 4
    for (int t = 0; t < Tk; ++t)
        acc = fmaf(p[t], vb[(size_t)t * DVk], acc);
    out_ctx[(size_t)b * DVk + d] = acc;
}

// ---------------------------------------------------------------------------
extern "C" void kernel_launch(void* const* d_in, const int* in_sizes, int n_in,
                              void* d_out, int out_size, void* d_ws, size_t ws_size,
                              hipStream_t stream) {
    (void)in_sizes; (void)n_in; (void)out_size; (void)ws_size;
    const float* query  = (const float*)d_in[0];
    const float* values = (const float*)d_in[1];
    const float* prevw  = (const float*)d_in[2];
    const float* Wq     = (const float*)d_in[3];
    const float* bq     = (const float*)d_in[4];
    const float* Wm     = (const float*)d_in[5];
    const float* bm     = (const float*)d_in[6];
    const float* Wv     = (const float*)d_in[7];
    const float* bv     = (const float*)d_in[8];
    const float* Wconv  = (const float*)d_in[9];
    const float* Wl     = (const float*)d_in[10];

    float* qws   = (float*)d_ws;                       // [64][128]
    float* sws   = qws + Bk * UNITSk;                  // [64][1024]
    float* ctx   = (float*)d_out;                      // [64][512]
    float* aw    = ctx + (size_t)Bk * DVk;             // [64][1000]

    lsa_qproj<<<Bk, UNITSk, 0, stream>>>(query, Wq, bq, bm, qws);

    dim3 g2((Tk + ROWS - 1) / ROWS, Bk);               // (32, 64)
    lsa_score<<<g2, 256, 0, stream>>>(values, prevw, Wm, Wv, bv, Wconv, Wl,
                                      qws, sws);

    lsa_softmax_ctx<<<Bk, 512, 0, stream>>>(sws, values, ctx, aw);
}